// RGCN_59210419143208
// MI455X (gfx1250) — compile-verified
//
#include <hip/hip_runtime.h>

#define N_NODES 50000
#define N_EDGES 800000
#define HDIM 128
#define N_RELS 24
#define LDS_STRIDE 132   // 128 + 4 pad: avoids 64-bank conflicts on row accesses
#define TILE_E 64        // edges (or nodes) per block = 4 WMMA row-tiles

typedef __attribute__((ext_vector_type(2))) float v2f;
typedef __attribute__((ext_vector_type(4))) float v4f;
typedef __attribute__((ext_vector_type(8))) float v8f;

// ---------------------------------------------------------------- utilities

__global__ void zero_f32(float* p, long n) {
    long i = (long)blockIdx.x * blockDim.x + threadIdx.x;
    long stride = (long)gridDim.x * blockDim.x;
    for (; i < n; i += stride) p[i] = 0.0f;
}

__global__ void hist_kernel(const int* __restrict__ rel, int* __restrict__ rel_count) {
    int e = blockIdx.x * blockDim.x + threadIdx.x;
    if (e < N_EDGES) atomicAdd(&rel_count[rel[e]], 1);
}

// 24 buckets: trivial serial scan. Also produce per-relation supertile offsets
// (supertiles of 64 edges) so msg_kernel can map blockIdx -> (r, supertile).
__global__ void scan_kernel(const int* __restrict__ rel_count,
                            int* __restrict__ rel_offset,
                            int* __restrict__ tile_start,
                            int* __restrict__ cursor) {
    if (threadIdx.x == 0 && blockIdx.x == 0) {
        int acc = 0, tacc = 0;
        for (int r = 0; r < N_RELS; ++r) {
            rel_offset[r] = acc;
            tile_start[r] = tacc;
            cursor[r]     = acc;
            int c = rel_count[r];
            acc  += c;
            tacc += (c + TILE_E - 1) / TILE_E;
        }
        rel_offset[N_RELS] = acc;
        tile_start[N_RELS] = tacc;
    }
}

__global__ void scatter_kernel(const int* __restrict__ rel, const int* __restrict__ dst,
                               int* __restrict__ cursor, int* __restrict__ perm,
                               float* __restrict__ deg) {
    int e = blockIdx.x * blockDim.x + threadIdx.x;
    if (e < N_EDGES) {
        int r = rel[e];
        int pos = atomicAdd(&cursor[r], 1);
        perm[pos] = e;
        atomicAdd(&deg[dst[e]], 1.0f);
    }
}

// ------------------------------------------------- main WMMA message kernel
// One block = 4 waves = 128 threads; processes 64 edges (4 row-tiles) of one
// relation. Wave w owns output columns [32w, 32w+32) for ALL 4 row-tiles, so
// every B fragment (from W[r]) is reused 4x: per K-step the loop body is
// 4 ds_load_b64 + 4 global_load_b32 + 8 v_wmma_f32_16x16x4_f32.
__global__ __launch_bounds__(128) void msg_kernel(
        const float* __restrict__ h, const float* __restrict__ W,
        const int* __restrict__ src, const int* __restrict__ dst,
        const int* __restrict__ perm, const int* __restrict__ rel_offset,
        const int* __restrict__ tile_start, float* __restrict__ agg) {
    __shared__ float Atile[TILE_E * LDS_STRIDE];   // 33.8 KB
    __shared__ int s_dst[TILE_E];
    __shared__ int s_meta[3];   // r, estart, ecount

    int t = threadIdx.x;
    int b = blockIdx.x;
    if (t == 0) {
        int ntiles = tile_start[N_RELS];
        if (b >= ntiles) {
            s_meta[0] = -1;
        } else {
            int r = 0;
            while (b >= tile_start[r + 1]) ++r;      // <=24 iterations
            int estart = rel_offset[r] + (b - tile_start[r]) * TILE_E;
            int ecount = rel_offset[r + 1] - estart;
            if (ecount > TILE_E) ecount = TILE_E;
            s_meta[0] = r; s_meta[1] = estart; s_meta[2] = ecount;
        }
    }
    __syncthreads();
    int r = s_meta[0];
    if (r < 0) return;                               // uniform across block
    int estart = s_meta[1];
    int ecount = s_meta[2];

    // Gather 64 source-node feature rows into LDS (zeros for padded rows).
    {
        int row = t >> 1;              // 64 rows x 2 threads
        int col = (t & 1) << 6;        // 64 floats per thread
        float* lp = Atile + row * LDS_STRIDE + col;
        if (row < ecount) {
            const float* hp = h + (long)src[perm[estart + row]] * HDIM + col;
            #pragma unroll
            for (int j = 0; j < 16; ++j)
                *(v4f*)(lp + 4 * j) = *(const v4f*)(hp + 4 * j);
        } else {
            #pragma unroll
            for (int j = 0; j < 64; ++j) lp[j] = 0.0f;
        }
        if (t < TILE_E) s_dst[t] = (t < ecount) ? dst[perm[estart + t]] : -1;
    }
    __syncthreads();

    int lane = t & 31;
    int wave = t >> 5;
    int half = lane >> 4;      // 0: lanes 0-15, 1: lanes 16-31
    int m    = lane & 15;      // A row (M) / B column (N) within 16x16 tile
    int n0   = wave << 5;
    int n1   = n0 + 16;
    const float* Wr = W + (long)r * HDIM * HDIM;

    v8f acc[4][2] = {};
    #pragma unroll 2
    for (int k = 0; k < HDIM; k += 4) {
        int kk = k + (half << 1);                    // K layout per ISA 7.12.2
        v2f b0, b1;
        b0.x = Wr[(long)kk       * HDIM + n0 + m];
        b0.y = Wr[(long)(kk + 1) * HDIM + n0 + m];
        b1.x = Wr[(long)kk       * HDIM + n1 + m];
        b1.y = Wr[(long)(kk + 1) * HDIM + n1 + m];
        #pragma unroll
        for (int tt = 0; tt < 4; ++tt) {
            v2f a = *(const v2f*)(Atile + (tt * 16 + m) * LDS_STRIDE + kk);
            acc[tt][0] = __builtin_amdgcn_wmma_f32_16x16x4_f32(
                false, a, false, b0, (short)0, acc[tt][0], false, false);
            acc[tt][1] = __builtin_amdgcn_wmma_f32_16x16x4_f32(
                false, a, false, b1, (short)0, acc[tt][1], false, false);
        }
    }

    // Scatter message rows into agg[dst] (L2-resident float atomics).
    #pragma unroll
    for (int tt = 0; tt < 4; ++tt) {
        #pragma unroll
        for (int v = 0; v < 8; ++v) {
            int row = tt * 16 + v + (half << 3);     // C/D layout: M = v + 8*half
            int nd = s_dst[row];
            if (nd >= 0) {
                atomicAdd(&agg[(long)nd * HDIM + n0 + m], acc[tt][0][v]);
                atomicAdd(&agg[(long)nd * HDIM + n1 + m], acc[tt][1][v]);
            }
        }
    }
}

// ------------------------------------------ fused self-loop + mean + relu
// out = relu(agg/max(deg,1) + h @ W_loop + bias); 64 nodes per block with the
// same 4x B-fragment reuse as msg_kernel.
__global__ __launch_bounds__(128) void finish_kernel(
        const float* __restrict__ h, const float* __restrict__ Wloop,
        const float* __restrict__ bias, const float* __restrict__ agg,
        const float* __restrict__ deg, float* __restrict__ out) {
    __shared__ float Atile[TILE_E * LDS_STRIDE];
    int t = threadIdx.x;
    int base = blockIdx.x * TILE_E;

    {
        int row = t >> 1;
        int col = (t & 1) << 6;
        int node = base + row;
        float* lp = Atile + row * LDS_STRIDE + col;
        if (node < N_NODES) {
            const float* hp = h + (long)node * HDIM + col;
            #pragma unroll
            for (int j = 0; j < 16; ++j)
                *(v4f*)(lp + 4 * j) = *(const v4f*)(hp + 4 * j);
        } else {
            #pragma unroll
            for (int j = 0; j < 64; ++j) lp[j] = 0.0f;
        }
    }
    __syncthreads();

    int lane = t & 31;
    int wave = t >> 5;
    int half = lane >> 4;
    int m    = lane & 15;
    int n0   = wave << 5;
    int n1   = n0 + 16;

    v8f acc[4][2] = {};
    #pragma unroll 2
    for (int k = 0; k < HDIM; k += 4) {
        int kk = k + (half << 1);
        v2f b0, b1;
        b0.x = Wloop[(long)kk       * HDIM + n0 + m];
        b0.y = Wloop[(long)(kk + 1) * HDIM + n0 + m];
        b1.x = Wloop[(long)kk       * HDIM + n1 + m];
        b1.y = Wloop[(long)(kk + 1) * HDIM + n1 + m];
        #pragma unroll
        for (int tt = 0; tt < 4; ++tt) {
            v2f a = *(const v2f*)(Atile + (tt * 16 + m) * LDS_STRIDE + kk);
            acc[tt][0] = __builtin_amdgcn_wmma_f32_16x16x4_f32(
                false, a, false, b0, (short)0, acc[tt][0], false, false);
            acc[tt][1] = __builtin_amdgcn_wmma_f32_16x16x4_f32(
                false, a, false, b1, (short)0, acc[tt][1], false, false);
        }
    }

    float bias0 = bias[n0 + m];
    float bias1 = bias[n1 + m];
    #pragma unroll
    for (int tt = 0; tt < 4; ++tt) {
        #pragma unroll
        for (int v = 0; v < 8; ++v) {
            int row = tt * 16 + v + (half << 3);
            int node = base + row;
            if (node < N_NODES) {
                float dg = deg[node];
                float inv = 1.0f / (dg > 1.0f ? dg : 1.0f);
                long o0 = (long)node * HDIM + n0 + m;
                long o1 = (long)node * HDIM + n1 + m;
                float val0 = agg[o0] * inv + acc[tt][0][v] + bias0;
                float val1 = agg[o1] * inv + acc[tt][1][v] + bias1;
                out[o0] = val0 > 0.0f ? val0 : 0.0f;
                out[o1] = val1 > 0.0f ? val1 : 0.0f;
            }
        }
    }
}

// ---------------------------------------------------------------- launcher

extern "C" void kernel_launch(void* const* d_in, const int* in_sizes, int n_in,
                              void* d_out, int out_size, void* d_ws, size_t ws_size,
                              hipStream_t stream) {
    (void)in_sizes; (void)n_in; (void)out_size; (void)ws_size;
    const float* h     = (const float*)d_in[0];
    const float* W     = (const float*)d_in[1];
    const float* Wloop = (const float*)d_in[2];
    const float* bias  = (const float*)d_in[3];
    const int*   src   = (const int*)d_in[4];
    const int*   dst   = (const int*)d_in[5];
    const int*   rel   = (const int*)d_in[6];
    float* out = (float*)d_out;

    // Workspace carve-up (~29 MB).
    float* agg       = (float*)d_ws;                       // N*D
    float* deg       = agg + (size_t)N_NODES * HDIM;       // N
    int*   perm      = (int*)(deg + N_NODES);              // E
    int*   rel_count = perm + N_EDGES;                     // R
    int*   rel_offset= rel_count + N_RELS;                 // R+1
    int*   tile_start= rel_offset + N_RELS + 1;            // R+1
    int*   cursor    = tile_start + N_RELS + 1;            // R

    zero_f32<<<2048, 256, 0, stream>>>(agg, (long)N_NODES * HDIM + N_NODES);
    zero_f32<<<1, 128, 0, stream>>>((float*)rel_count, 4 * N_RELS + 2);

    hist_kernel<<<(N_EDGES + 255) / 256, 256, 0, stream>>>(rel, rel_count);
    scan_kernel<<<1, 32, 0, stream>>>(rel_count, rel_offset, tile_start, cursor);
    scatter_kernel<<<(N_EDGES + 255) / 256, 256, 0, stream>>>(rel, dst, cursor, perm, deg);

    int max_tiles = N_EDGES / TILE_E + N_RELS;   // >= sum_r ceil(count_r/64)
    msg_kernel<<<max_tiles, 128, 0, stream>>>(h, W, src, dst, perm,
                                              rel_offset, tile_start, agg);
    finish_kernel<<<(N_NODES + TILE_E - 1) / TILE_E, 128, 0, stream>>>(
        h, Wloop, bias, agg, deg, out);
}